// POELayer_76270029243109
// MI455X (gfx1250) — compile-verified
//
#include <hip/hip_runtime.h>
#include <hip/hip_bf16.h>
#include <math.h>

#define NJ 24
#define BLOCK 256

typedef __attribute__((ext_vector_type(2))) float v2f;
typedef __attribute__((ext_vector_type(8))) float v8f;

#if defined(__gfx1250__) && __has_builtin(__builtin_amdgcn_wmma_f32_16x16x4_f32)
#define USE_WMMA 1
#endif

#if defined(__gfx1250__) && __has_builtin(__builtin_amdgcn_global_load_async_to_lds_b128) && \
    __has_builtin(__builtin_amdgcn_s_wait_asynccnt)
#define USE_ASYNC 1
typedef __attribute__((ext_vector_type(4))) int v4i;
typedef __attribute__((address_space(1))) v4i gv4i;  // global int4*
typedef __attribute__((address_space(3))) v4i sv4i;  // LDS int4*
#endif

// exp_se3 of xi = s * e6, returned as affine 3x4 [R | p] (bottom row implicitly 0,0,0,1).
// Matches the reference's branchless small-angle selects in structure.
__device__ __forceinline__ void exp_se3_affine(float s, const float* e6, float E[3][4]) {
  float wx = s * e6[0], wy = s * e6[1], wz = s * e6[2];
  float vx = s * e6[3], vy = s * e6[4], vz = s * e6[5];
  float th2 = fmaf(wx, wx, fmaf(wy, wy, wz * wz));
  bool sm = th2 < 1e-12f;
  float th2s = sm ? 1.0f : th2;
  float th = sqrtf(th2s);
  float sn, cs;
  sincosf(th, &sn, &cs);
  float A = sm ? fmaf(th2, -1.0f / 6.0f, 1.0f) : sn / th;
  float B = sm ? fmaf(th2, -1.0f / 24.0f, 0.5f) : (1.0f - cs) / th2s;
  float C = sm ? fmaf(th2, -1.0f / 120.0f, 1.0f / 6.0f) : (th - sn) / (th2s * th);
  float xx = wx * wx, yy = wy * wy, zz = wz * wz;
  float xy = wx * wy, xz = wx * wz, yz = wy * wz;
  // R = I + A*W + B*(w w^T - th2 I)
  E[0][0] = fmaf(B, xx - th2, 1.0f);
  E[0][1] = fmaf(B, xy, -A * wz);
  E[0][2] = fmaf(B, xz,  A * wy);
  E[1][0] = fmaf(B, xy,  A * wz);
  E[1][1] = fmaf(B, yy - th2, 1.0f);
  E[1][2] = fmaf(B, yz, -A * wx);
  E[2][0] = fmaf(B, xz, -A * wy);
  E[2][1] = fmaf(B, yz,  A * wx);
  E[2][2] = fmaf(B, zz - th2, 1.0f);
  // V = I + B*W + C*(w w^T - th2 I);  p = V v
  float V00 = fmaf(C, xx - th2, 1.0f), V01 = fmaf(C, xy, -B * wz), V02 = fmaf(C, xz, B * wy);
  float V10 = fmaf(C, xy, B * wz), V11 = fmaf(C, yy - th2, 1.0f), V12 = fmaf(C, yz, -B * wx);
  float V20 = fmaf(C, xz, -B * wy), V21 = fmaf(C, yz, B * wx), V22 = fmaf(C, zz - th2, 1.0f);
  E[0][3] = fmaf(V00, vx, fmaf(V01, vy, V02 * vz));
  E[1][3] = fmaf(V10, vx, fmaf(V11, vy, V12 * vz));
  E[2][3] = fmaf(V20, vx, fmaf(V21, vy, V22 * vz));
}

__global__ __launch_bounds__(BLOCK) void poe_kernel(const float* __restrict__ x,
                                                    const float* __restrict__ eta,
                                                    const float* __restrict__ Mse3,
                                                    float* __restrict__ out) {
  __shared__ float xs[BLOCK * (NJ + 1)];   // padded stride 25 -> conflict-free row reads
  __shared__ float etas[NJ * 6];
  __shared__ float Tbuf[BLOCK * 16];       // full 4x4 per thread, row-major
  __shared__ float EM[16];                 // exp_se3(M), full 4x4 row-major

  const int tid = threadIdx.x;
  const int blockBase = blockIdx.x * BLOCK;
  const float* gx = x + (size_t)blockBase * NJ;

#ifdef USE_ASYNC
  // ---- async global->LDS staging (ASYNCcnt path), 6 coalesced b128 per thread ----
  // float4 index f covers 1536 groups; group f -> row b=f/6, cols [4*(f%6) .. +3].
  // Groups never straddle the padded 25-float LDS row; global side 16B aligned, LDS dst only 4B aligned (100 B row stride).
  #pragma unroll
  for (int i = 0; i < 6; ++i) {
    int f = tid + BLOCK * i;
    int b = f / 6, jg = f - b * 6;
    const float* gsrc = gx + (size_t)f * 4;
    float* ldst = &xs[b * (NJ + 1) + jg * 4];
    __builtin_amdgcn_global_load_async_to_lds_b128((gv4i*)(void*)gsrc, (sv4i*)(void*)ldst, 0, 0);
  }
  if (tid < (NJ * 6) / 4) {
    __builtin_amdgcn_global_load_async_to_lds_b128((gv4i*)(void*)(eta + tid * 4),
                                                   (sv4i*)(void*)&etas[tid * 4], 0, 0);
  }
  __builtin_amdgcn_s_wait_asynccnt(0);
#else
  for (int i = tid; i < BLOCK * NJ; i += BLOCK) {
    float v = gx[i];
    int bl = i / NJ, jj = i - bl * NJ;
    xs[bl * (NJ + 1) + jj] = v;
  }
  for (int i = tid; i < NJ * 6; i += BLOCK) etas[i] = eta[i];
#endif

  if (tid == 0) {
    float Em[3][4];
    exp_se3_affine(1.0f, Mse3, Em);
    for (int r = 0; r < 3; ++r)
      for (int c = 0; c < 4; ++c) EM[r * 4 + c] = Em[r][c];
    EM[12] = 0.0f; EM[13] = 0.0f; EM[14] = 0.0f; EM[15] = 1.0f;
  }
  __syncthreads();

  // ---- serial product-of-exponentials chain, fully in registers ----
  float t[3][4] = {{1.f, 0.f, 0.f, 0.f}, {0.f, 1.f, 0.f, 0.f}, {0.f, 0.f, 1.f, 0.f}};
  const float* xrow = &xs[tid * (NJ + 1)];
  for (int j = 0; j < NJ; ++j) {
    float E[3][4];
    exp_se3_affine(xrow[j], &etas[j * 6], E);
    #pragma unroll
    for (int i = 0; i < 3; ++i) {
      float a0 = t[i][0], a1 = t[i][1], a2 = t[i][2];
      t[i][0] = fmaf(a0, E[0][0], fmaf(a1, E[1][0], a2 * E[2][0]));
      t[i][1] = fmaf(a0, E[0][1], fmaf(a1, E[1][1], a2 * E[2][1]));
      t[i][2] = fmaf(a0, E[0][2], fmaf(a1, E[1][2], a2 * E[2][2]));
      t[i][3] = fmaf(a0, E[0][3], fmaf(a1, E[1][3], fmaf(a2, E[2][3], t[i][3])));
    }
  }

#ifdef USE_WMMA
  // ---- epilogue: out_b = T_b @ EM as a batched GEMM via V_WMMA_F32_16X16X4_F32 ----
  // out^T = EM^T @ T^T:  A(16x4)[m,k] = EM[k, m%4] (rows replicated 4x),
  //                      B(4x16)[k, 4e+j] = T_e[j, k]  (4 elements per WMMA).
  // C[i, 4e+j] = out_e[j, i%4]  -> C vgprs 0..3 on lanes 0..15 are contiguous
  // output rows -> one fully-coalesced float4 store per active lane.
  {
    int tb = tid * 16;
    Tbuf[tb + 0] = t[0][0]; Tbuf[tb + 1] = t[0][1]; Tbuf[tb + 2]  = t[0][2]; Tbuf[tb + 3]  = t[0][3];
    Tbuf[tb + 4] = t[1][0]; Tbuf[tb + 5] = t[1][1]; Tbuf[tb + 6]  = t[1][2]; Tbuf[tb + 7]  = t[1][3];
    Tbuf[tb + 8] = t[2][0]; Tbuf[tb + 9] = t[2][1]; Tbuf[tb + 10] = t[2][2]; Tbuf[tb + 11] = t[2][3];
    Tbuf[tb + 12] = 0.0f;   Tbuf[tb + 13] = 0.0f;   Tbuf[tb + 14] = 0.0f;    Tbuf[tb + 15] = 1.0f;
  }
  __syncthreads();

  const int lane = tid & 31;
  const int waveId = tid >> 5;
  const int m = lane & 15;            // column (lanes 0-15) / row index (A layout)
  const int k0 = (lane < 16) ? 0 : 2; // K-pair held by this half-wave
  const int colE = m >> 2;            // element within WMMA group
  const int jj = m & 3;               // output row index carried by this column

  // A is loop-invariant: A[m,k0] = EM[k0, m%4], A[m,k0+1] = EM[k0+1, m%4]
  v2f a;
  a.x = EM[k0 * 4 + (m & 3)];
  a.y = EM[(k0 + 1) * 4 + (m & 3)];

  const float* wT = &Tbuf[waveId * 32 * 16];
  #pragma unroll
  for (int g = 0; g < 8; ++g) {
    const float* src = &wT[(4 * g + colE) * 16 + jj * 4 + k0]; // T_e[jj, k0], T_e[jj, k0+1]
    v2f b; b.x = src[0]; b.y = src[1];
    v8f cacc = {};
    cacc = __builtin_amdgcn_wmma_f32_16x16x4_f32(
        /*neg_a=*/false, a, /*neg_b=*/false, b,
        /*c_mod=*/(short)0, cacc, /*reuse_a=*/false, /*reuse_b=*/false);
    if (lane < 16) {
      int e = lane >> 2, j = lane & 3;
      size_t ob = ((size_t)(blockBase + waveId * 32 + 4 * g + e)) * 16 + (size_t)j * 4;
      float4 r; r.x = cacc[0]; r.y = cacc[1]; r.z = cacc[2]; r.w = cacc[3];
      *reinterpret_cast<float4*>(out + ob) = r;
    }
  }
#else
  // ---- fallback VALU epilogue: out = T @ EM per thread ----
  float o[4][4];
  #pragma unroll
  for (int i = 0; i < 3; ++i) {
    #pragma unroll
    for (int c = 0; c < 4; ++c) {
      float acc = (c == 3) ? t[i][3] : 0.0f;
      acc = fmaf(t[i][0], EM[0 * 4 + c], acc);
      acc = fmaf(t[i][1], EM[1 * 4 + c], acc);
      acc = fmaf(t[i][2], EM[2 * 4 + c], acc);
      o[i][c] = acc;
    }
  }
  o[3][0] = 0.0f; o[3][1] = 0.0f; o[3][2] = 0.0f; o[3][3] = 1.0f;
  float* ob = out + (size_t)(blockBase + tid) * 16;
  #pragma unroll
  for (int i = 0; i < 4; ++i) {
    float4 r; r.x = o[i][0]; r.y = o[i][1]; r.z = o[i][2]; r.w = o[i][3];
    *reinterpret_cast<float4*>(ob + i * 4) = r;
  }
#endif
}

extern "C" void kernel_launch(void* const* d_in, const int* in_sizes, int n_in,
                              void* d_out, int out_size, void* d_ws, size_t ws_size,
                              hipStream_t stream) {
  (void)in_sizes; (void)n_in; (void)d_ws; (void)ws_size;
  const float* x    = (const float*)d_in[0];   // (262144, 24)
  const float* eta  = (const float*)d_in[1];   // (24, 6)
  const float* Mse3 = (const float*)d_in[2];   // (1, 6)
  float* out = (float*)d_out;                  // (262144, 4, 4)
  const int nBatch = in_sizes[0] / NJ;
  dim3 grid(nBatch / BLOCK), block(BLOCK);
  poe_kernel<<<grid, block, 0, stream>>>(x, eta, Mse3, out);
  (void)out_size;
}